// fhgc_57174604644692
// MI455X (gfx1250) — compile-verified
//
#include <hip/hip_runtime.h>

// ---------------------------------------------------------------------------
// Fused TSM-style temporal-difference module for gfx1250 (MI455X, wave32).
//
//  Stage 1 (per t = 0..6): depthwise 3x3 dilated(2) conv via
//      V_WMMA_F32_16X16X4_F32 (exact fp32 matrix path):
//      16x16 output tile = sum over 3 kernel rows of
//      A(16x20 shifted x columns) x B(20x16 banded Toeplitz of the 3 taps),
//      split into five K=4 WMMA accumulations.
//      Tiles overlap at the right/bottom edges (rows {0,12}, cols
//      {0,16,32,40}) so every C-element store is in-range -> no exec masking.
//      xpad is double-buffered and the fill is software-pipelined by hand:
//      both of t+1's global_load_b128 issue BEFORE the WMMA block (branch-free
//      via index clamping: the clamped duplicate rewrites the same LDS word),
//      and the matching ds_store_b128s land AFTER it, so the loadcnt waits
//      are covered by matrix compute.
//  Stage 2: temporal combine out[d] = sum_k w_t[k] * tgt[2d+k] where
//      tgt[m] = agg[m/3] (m%3==0) | x[m/3+1] (m%3==1) | agg[m/3+2] (m%3==2),
//      fully unrolled over d (source planes resolved at compile time),
//      float4-vectorized, 2 iterations per thread.
//
//  One workgroup = one (batch, channel, row-half). agg planes live in LDS.
// ---------------------------------------------------------------------------

typedef float v2f __attribute__((ext_vector_type(2)));
typedef float v8f __attribute__((ext_vector_type(8)));

#define TSEG   8
#define CCH    256
#define HH     56
#define WW     56
#define PLANE  (HH * WW)        // 3136
#define RH     28               // rows per half-plane block
#define PRROWS 32               // padded rows: top pad 2  (-2 .. +29)
#define PCCOLS 68               // padded cols: left pad 4 (-4 .. +63), pitch 272B
#define PADT   2
#define PADL   4
#define NFILLROWS 30            // valid x rows touching this half (+/-2 halo)
#define NV4ROW    14            // 56 floats = 14 float4 per row
#define NFILL4    (NFILLROWS * NV4ROW)   // 420 float4 per fill

__global__ __launch_bounds__(256)
void tsm_fused_kernel(const float* __restrict__ x,
                      const float* __restrict__ w_spa,
                      const float* __restrict__ w_t,
                      float* __restrict__ out)
{
    __shared__ __align__(16) float xpad[2][PRROWS * PCCOLS]; // 2 x 8,704 B
    __shared__ __align__(16) float aggbuf[7][RH * WW];       // 43,904 B

    const int blk  = blockIdx.x;
    const int sh   = blk & 1;                 // row half: rows [sh*28, sh*28+28)
    const int ch   = (blk >> 1) & (CCH - 1);
    const int bb   = blk >> 9;
    const int tid  = threadIdx.x;
    const int lane = tid & 31;
    const int wave = tid >> 5;
    const int r0   = sh * RH;

    // --- weights (uniform per block -> scalarized by compiler) -------------
    float ws[9];
#pragma unroll
    for (int i = 0; i < 9; ++i) ws[i] = w_spa[ch * 9 + i];
    const float kw0 = w_t[ch * 3 + 0];
    const float kw1 = w_t[ch * 3 + 1];
    const float kw2 = w_t[ch * 3 + 2];

    // --- Toeplitz B fragments for V_WMMA_F32_16X16X4_F32 -------------------
    // B is 4x16 (KxN). Lane l, vgpr v holds element (K = v + 2*(l/16), N = l%16).
    // Chunk q covers band rows p = 4q+K of the 20x16 Toeplitz:
    //   B[p, n] = ws[ky*3 + kx]  iff  p - n == 2*kx  (dilated taps at 0,2,4)
    const int n     = lane & 15;
    const int khalf = (lane >> 4) << 1;       // 0 or 2
    v2f Bm[3][5];
#pragma unroll
    for (int ky = 0; ky < 3; ++ky) {
#pragma unroll
        for (int q = 0; q < 5; ++q) {
#pragma unroll
            for (int v = 0; v < 2; ++v) {
                const int diff = 4 * q + khalf + v - n;
                float bv = 0.0f;
                if (diff == 0)      bv = ws[ky * 3 + 0];
                else if (diff == 2) bv = ws[ky * 3 + 1];
                else if (diff == 4) bv = ws[ky * 3 + 2];
                Bm[ky][q][v] = bv;
            }
        }
    }

    // --- per-wave 16x16 tile; overlapping tiles cover 28x56 exactly --------
    const int ti = wave >> 2;                 // 0..1
    const int tj = wave & 3;                  // 0..3
    const int i0 = ti * 12;                   // {0, 12}  (rows 0..15 / 12..27)
    const int j0 = (tj < 3) ? tj * 16 : 40;   // {0,16,32,40} (cols ..55)
    const int m  = lane & 15;                 // A-matrix row inside tile

    const float* xplane0 = x + ((size_t)bb * TSEG * CCH + ch) * PLANE;

    // --- fill geometry (t-invariant): 420 float4 = 2 per thread ------------
    // src float4 index is simply the flat element index e (row*14 + c4).
    // Element 1 is clamped to element 0 for tid >= 164: then src1==src0 and
    // dst1==dst0, so the unconditional duplicate store is benign. This keeps
    // the whole fill branch-free so both loads hoist above the WMMA block.
    const int gr0 = (r0 - PADT < 0) ? 0 : (r0 - PADT);   // first valid x row
    const int pr0 = gr0 - r0 + PADT;                     // its padded row
    const int e1  = (tid + 256 < NFILL4) ? (tid + 256) : tid;
    const int dst0 = (pr0 + tid / NV4ROW) * (PCCOLS / 4) + (PADL / 4)
                     + (tid - (tid / NV4ROW) * NV4ROW);
    const int dst1 = (pr0 + e1 / NV4ROW) * (PCCOLS / 4) + (PADL / 4)
                     + (e1 - (e1 / NV4ROW) * NV4ROW);

    // --- one-time zero of both padded buffers (border is t-invariant) ------
    {
        float4* z = reinterpret_cast<float4*>(&xpad[0][0]);
        const float4 zz = make_float4(0.f, 0.f, 0.f, 0.f);
        z[tid      ] = zz;
        z[tid + 256] = zz;
        z[tid + 512] = zz;
        z[tid + 768] = zz;
        if (tid < 2 * (PRROWS * PCCOLS) / 4 - 1024) z[tid + 1024] = zz;
    }
    __syncthreads();

    // --- prologue: fill buffer 0 with plane t=0 ----------------------------
    {
        const float4* src = reinterpret_cast<const float4*>(xplane0 + gr0 * WW);
        float4* dst = reinterpret_cast<float4*>(&xpad[0][0]);
        dst[dst0] = src[tid];
        dst[dst1] = src[e1];
    }
    __syncthreads();

    // ======================= Stage 1: agg planes ===========================
    for (int t = 0; t < 7; ++t) {
        // ---- issue next plane's global loads FIRST (no wait yet) ----------
        float4 nbuf0, nbuf1;
        if (t < 6) {
            const float4* src = reinterpret_cast<const float4*>(
                xplane0 + (size_t)(t + 1) * CCH * PLANE + gr0 * WW);
            nbuf0 = src[tid];
            nbuf1 = src[e1];
        }

        const float* cur = &xpad[t & 1][0];

        // 15 fp32 WMMAs accumulate the full dilated 3x3 conv for this tile;
        // the outstanding global loads complete underneath them.
        v8f acc = {0.f, 0.f, 0.f, 0.f, 0.f, 0.f, 0.f, 0.f};
#pragma unroll
        for (int ky = 0; ky < 3; ++ky) {
            // A element (M=m, K) = x[row i0+m+2(ky-1), col j0-2+4q+K]
            // padded: row + PADT, col + PADL
            const float* arow = &cur[(i0 + m + 2 * ky) * PCCOLS + j0 + 2 + khalf];
#pragma unroll
            for (int q = 0; q < 5; ++q) {
                v2f a;
                a.x = arow[4 * q];
                a.y = arow[4 * q + 1];
                acc = __builtin_amdgcn_wmma_f32_16x16x4_f32(
                          false, a, false, Bm[ky][q], (short)0, acc,
                          false, false);
            }
        }

        // C layout: vgpr r, lane l -> (row i0 + r + 8*(l/16), col j0 + l%16)
        // Overlapped tiling => always in range; duplicate writes identical.
        const int col   = j0 + n;
        const int rbase = i0 + ((lane >> 4) << 3);
        float* abase = &aggbuf[t][rbase * WW + col];
#pragma unroll
        for (int r = 0; r < 8; ++r) {
            abase[r * WW] = acc[r];
        }

        // ---- now commit next plane's data to the other xpad buffer -------
        // (branch-free: clamped duplicate rewrites the same word)
        if (t < 6) {
            float4* dst = reinterpret_cast<float4*>(&xpad[(t + 1) & 1][0]);
            dst[dst0] = nbuf0;
            dst[dst1] = nbuf1;
        }
        __syncthreads();   // fill(t+1) done, agg[t] visible, xpad[t&1] free
    }

    // ======================= Stage 2: temporal combine =====================
    // Fully unrolled over d: source planes (one global-x, two LDS agg) are
    // compile-time; 392 float4 per (d,half) plane = 2 iterations per thread.
    const float kw[3] = {kw0, kw1, kw2};
    const int NV4_PER_D = (RH * WW) / 4;            // 392
    const bool has2 = (tid + 256) < NV4_PER_D;      // tid < 136
    float* outbase = out + ((size_t)(bb * TSEG) * CCH + ch) * PLANE + r0 * WW;
#pragma unroll
    for (int d = 0; d < TSEG; ++d) {
#pragma unroll
        for (int h = 0; h < 2; ++h) {
            const int rj4 = tid + 256 * h;
            if (h == 0 || has2) {
                float4 o = make_float4(0.f, 0.f, 0.f, 0.f);
#pragma unroll
                for (int k = 0; k < 3; ++k) {
                    const int mm = 2 * d + k;       // compile-time
                    const int tt = mm / 3;
                    const int ss = mm - 3 * tt;
                    float4 v;
                    if (ss == 1) {
                        const float4* xv = reinterpret_cast<const float4*>(
                            xplane0 + (size_t)(tt + 1) * CCH * PLANE + r0 * WW);
                        v = xv[rj4];
                    } else {
                        const float4* av = reinterpret_cast<const float4*>(
                            &aggbuf[ss == 0 ? tt : tt + 2][0]);
                        v = av[rj4];
                    }
                    o.x = fmaf(kw[k], v.x, o.x);
                    o.y = fmaf(kw[k], v.y, o.y);
                    o.z = fmaf(kw[k], v.z, o.z);
                    o.w = fmaf(kw[k], v.w, o.w);
                }
                float4* ov = reinterpret_cast<float4*>(
                    outbase + (size_t)d * CCH * PLANE);
                ov[rj4] = o;
            }
        }
    }
}

extern "C" void kernel_launch(void* const* d_in, const int* in_sizes, int n_in,
                              void* d_out, int out_size, void* d_ws, size_t ws_size,
                              hipStream_t stream)
{
    (void)in_sizes; (void)n_in; (void)out_size; (void)d_ws; (void)ws_size;
    const float* x     = (const float*)d_in[0];
    const float* w_spa = (const float*)d_in[1];
    const float* w_t   = (const float*)d_in[2];
    float* out = (float*)d_out;

    const int nblocks = 8 * CCH * 2;   // batch * channels * row-halves = 4096
    tsm_fused_kernel<<<nblocks, 256, 0, stream>>>(x, w_spa, w_t, out);
}